// PCTransformer_9689446219929
// MI455X (gfx1250) — compile-verified
//
#include <hip/hip_runtime.h>
#include <math.h>

// ---------------------------------------------------------------------------
// CDNA5 (gfx1250) WMMA f32 path: V_WMMA_F32_16X16X4_F32, D = A(16x4) x B(4x16) + C
// ---------------------------------------------------------------------------
typedef __attribute__((ext_vector_type(2))) float v2f;
typedef __attribute__((ext_vector_type(8))) float v8f;

static __device__ __forceinline__ v8f wmma4(v2f a, v2f b, v8f c) {
  // 8 args: (neg_a, A, neg_b, B, c_mod, C, reuse_a, reuse_b)
  return __builtin_amdgcn_wmma_f32_16x16x4_f32(false, a, false, b, (short)0, c,
                                               false, false);
}

// ---------------------------------------------------------------------------
// Generic batched GEMM:  D = alpha * opA(A) @ opB(B) + bias[n] + beta * C
// opA(A) is MxK, opB(B) is KxN.  Batch index z -> (pb = z / Hb, ph = z % Hb),
// pointer offsets = pb*s?b + ph*s?h.  Requires M%64==0, N%64==0, K%32==0.
// ---------------------------------------------------------------------------
struct GemmP {
  const float *A, *B, *bias, *C;
  float *Dst;
  long long sAb, sAh, sBb, sBh, sCb, sCh, sDb, sDh;
  int M, N, K, lda, ldb, ldc, ldd, transA, transB, Hb;
  float alpha, beta;
};

__global__ __launch_bounds__(128) void k_gemm(GemmP p) {
  const int z  = blockIdx.z;
  const int pb = z / p.Hb;
  const int ph = z - pb * p.Hb;
  const float* A = p.A + (long long)pb * p.sAb + (long long)ph * p.sAh;
  const float* B = p.B + (long long)pb * p.sBb + (long long)ph * p.sBh;
  const float* C = p.C ? (p.C + (long long)pb * p.sCb + (long long)ph * p.sCh) : nullptr;
  float* Dst = p.Dst + (long long)pb * p.sDb + (long long)ph * p.sDh;

  const int m0 = blockIdx.y * 64;
  const int n0 = blockIdx.x * 64;

  __shared__ float sA[64 * 33];  // [m][k], padded
  __shared__ float sB[32 * 65];  // [k][n], padded

  v8f acc[2][2];
#pragma unroll
  for (int a = 0; a < 2; ++a)
#pragma unroll
    for (int b = 0; b < 2; ++b) acc[a][b] = (v8f){0.f,0.f,0.f,0.f,0.f,0.f,0.f,0.f};

  const int tid  = threadIdx.x;
  const int lane = tid & 31;
  const int w    = tid >> 5;
  const int mW   = (w >> 1) * 32;
  const int nW   = (w & 1) * 32;
  const int ml   = lane & 15;
  const int kh   = (lane >> 4) << 1;  // 0 for lanes 0-15, 2 for lanes 16-31

  for (int kk = 0; kk < p.K; kk += 32) {
    // ---- stage A tile -> sA[m][k] ----
    if (!p.transA) {
      for (int i = tid; i < 64 * 32; i += 128) {
        int m = i >> 5, k = i & 31;
        sA[m * 33 + k] = A[(size_t)(m0 + m) * p.lda + kk + k];
      }
    } else {
      for (int i = tid; i < 64 * 32; i += 128) {
        int m = i & 63, k = i >> 6;
        sA[m * 33 + k] = A[(size_t)(kk + k) * p.lda + m0 + m];
      }
    }
    // ---- stage B tile -> sB[k][n] ----
    if (!p.transB) {
      for (int i = tid; i < 32 * 64; i += 128) {
        int n = i & 63, k = i >> 6;
        sB[k * 65 + n] = B[(size_t)(kk + k) * p.ldb + n0 + n];
      }
    } else {
      for (int i = tid; i < 32 * 64; i += 128) {
        int k = i & 31, n = i >> 5;
        sB[k * 65 + n] = B[(size_t)(n0 + n) * p.ldb + kk + k];
      }
    }
    // prefetch next K tile (global_prefetch_b8)
    if (kk + 32 < p.K) {
      const float* pa = p.transA ? &A[(size_t)(kk + 32) * p.lda + m0 + (tid & 63)]
                                 : &A[(size_t)(m0 + (tid >> 1)) * p.lda + kk + 32];
      const float* pbp = p.transB ? &B[(size_t)(n0 + (tid >> 1)) * p.ldb + kk + 32]
                                  : &B[(size_t)(kk + 32 + (tid >> 2)) * p.ldb + n0];
      __builtin_prefetch(pa, 0, 0);
      __builtin_prefetch(pbp, 0, 0);
    }
    __syncthreads();

#pragma unroll
    for (int k = 0; k < 32; k += 4) {
      v2f a0 = {sA[(mW + ml) * 33 + k + kh],      sA[(mW + ml) * 33 + k + kh + 1]};
      v2f a1 = {sA[(mW + 16 + ml) * 33 + k + kh], sA[(mW + 16 + ml) * 33 + k + kh + 1]};
      v2f b0 = {sB[(k + kh) * 65 + nW + ml],      sB[(k + kh + 1) * 65 + nW + ml]};
      v2f b1 = {sB[(k + kh) * 65 + nW + 16 + ml], sB[(k + kh + 1) * 65 + nW + 16 + ml]};
      acc[0][0] = wmma4(a0, b0, acc[0][0]);
      acc[0][1] = wmma4(a0, b1, acc[0][1]);
      acc[1][0] = wmma4(a1, b0, acc[1][0]);
      acc[1][1] = wmma4(a1, b1, acc[1][1]);
    }
    __syncthreads();
  }

  // ---- epilogue: D = alpha*acc + bias + beta*C ----
  const int mhalf = (lane >> 4) * 8;
#pragma unroll
  for (int im = 0; im < 2; ++im) {
#pragma unroll
    for (int in = 0; in < 2; ++in) {
      const int ncol = n0 + nW + in * 16 + ml;
      const float bv = p.bias ? p.bias[ncol] : 0.f;
#pragma unroll
      for (int r = 0; r < 8; ++r) {
        const int mrow = m0 + mW + im * 16 + mhalf + r;
        float v = p.alpha * acc[im][in][r] + bv;
        if (C) v += p.beta * C[(size_t)mrow * p.ldc + ncol];
        Dst[(size_t)mrow * p.ldd + ncol] = v;
      }
    }
  }
}

// ---------------------------------------------------------------------------
// Elementwise / reduction kernels
// ---------------------------------------------------------------------------
static __device__ __forceinline__ float block_reduce_sum(float v, float* red) {
  const int tid = threadIdx.x;
  red[tid] = v;
  __syncthreads();
  for (int o = blockDim.x >> 1; o > 0; o >>= 1) {
    if (tid < o) red[tid] += red[tid + o];
    __syncthreads();
  }
  float r = red[0];
  __syncthreads();
  return r;
}

__global__ void k_zero(float* p, int n) {
  int i = blockIdx.x * blockDim.x + threadIdx.x;
  if (i < n) p[i] = 0.f;
}

// E = X - Y, energy += sum(E^2)
__global__ __launch_bounds__(256) void k_diff_energy(const float* X, const float* Y,
                                                     float* E, float* en, int n) {
  __shared__ float red[256];
  float s = 0.f;
  for (int i = blockIdx.x * 256 + threadIdx.x; i < n; i += gridDim.x * 256) {
    float e = X[i] - Y[i];
    E[i] = e;
    s += e * e;
  }
  float t = block_reduce_sum(s, red);
  if (threadIdx.x == 0) atomicAdd(en, t);
}

// causal softmax over each row of P [(B*H*S) x S], in place
__global__ __launch_bounds__(256) void k_softmax_causal(float* P, int S) {
  const int row = blockIdx.x;
  const int iq  = row % S;
  float* pr = P + (size_t)row * S;
  __shared__ float red[256];
  const int tid = threadIdx.x;
  float m = -3.4028235e38f;
  for (int j = tid; j < S; j += 256)
    if (j <= iq) m = fmaxf(m, pr[j]);
  red[tid] = m;
  __syncthreads();
  for (int o = 128; o > 0; o >>= 1) {
    if (tid < o) red[tid] = fmaxf(red[tid], red[tid + o]);
    __syncthreads();
  }
  m = red[0];
  __syncthreads();
  float s = 0.f;
  for (int j = tid; j < S; j += 256) {
    float e = (j <= iq) ? expf(pr[j] - m) : 0.f;
    pr[j] = e;
    s += e;
  }
  float tot = block_reduce_sum(s, red);
  const float inv = 1.f / tot;
  for (int j = tid; j < S; j += 256) pr[j] *= inv;
}

// dP <- P * (dP - sum(P*dP)) * scale   (softmax VJP; masked entries have P==0)
__global__ __launch_bounds__(256) void k_softmax_bwd(float* dP, const float* P,
                                                     float scale, int S) {
  const int row = blockIdx.x;
  const float* pr = P + (size_t)row * S;
  float* dr = dP + (size_t)row * S;
  __shared__ float red[256];
  float t = 0.f;
  for (int j = threadIdx.x; j < S; j += 256) t += pr[j] * dr[j];
  float T = block_reduce_sum(t, red);
  for (int j = threadIdx.x; j < S; j += 256) dr[j] = pr[j] * (dr[j] - T) * scale;
}

// exact GELU fwd+bwd: eps = tgt - gelu(u); G = eps * gelu'(u); energy += eps^2
__global__ __launch_bounds__(256) void k_gelu_bwd(const float* u, const float* tgt,
                                                  float* G, float* en, int n) {
  __shared__ float red[256];
  float s = 0.f;
  for (int i = blockIdx.x * 256 + threadIdx.x; i < n; i += gridDim.x * 256) {
    float x   = u[i];
    float cdf = 0.5f * (1.f + erff(x * 0.70710678118654752f));
    float ge  = x * cdf;
    float eps = tgt[i] - ge;
    float dg  = cdf + x * 0.39894228040143268f * expf(-0.5f * x * x);
    G[i] = eps * dg;
    s += eps * eps;
  }
  float t = block_reduce_sum(s, red);
  if (threadIdx.x == 0) atomicAdd(en, t);
}

// Pred[row,v] <- onehot(tgt[row])[v] - Pred[row,v]; energy += eps^2
__global__ __launch_bounds__(256) void k_eps_out(float* Pred, const int* tgt,
                                                 float* en, int V) {
  const int row = blockIdx.y;
  const int v   = blockIdx.x * 256 + threadIdx.x;
  const int tv  = tgt[row];
  const size_t idx = (size_t)row * V + v;
  float e = ((v == tv) ? 1.f : 0.f) - Pred[idx];
  Pred[idx] = e;
  __shared__ float red[256];
  float t = block_reduce_sum(e * e, red);
  if (threadIdx.x == 0) atomicAdd(en, t);
}

// energy += sum( (xq0 - (word_emb[ids]+pos_emb))^2 )
__global__ __launch_bounds__(256) void k_eps_embed(const float* xq0, const int* ids,
                                                   const float* we, const float* pe,
                                                   float* en, int Bb, int S, int D) {
  __shared__ float red[256];
  const int n = Bb * S * D;
  float s = 0.f;
  for (int i = blockIdx.x * 256 + threadIdx.x; i < n; i += gridDim.x * 256) {
    int d = i % D;
    int r = i / D;
    int ss = r % S;
    int b  = r / S;
    float pred = we[(size_t)ids[b * S + ss] * D + d] + pe[(size_t)ss * D + d];
    float e = xq0[i] - pred;
    s += e * e;
  }
  float t = block_reduce_sum(s, red);
  if (threadIdx.x == 0) atomicAdd(en, t);
}

__global__ void k_copy1(float* dst, const float* src) {
  if (threadIdx.x == 0 && blockIdx.x == 0) dst[0] = src[0];
}

// ---------------------------------------------------------------------------
// Host-side helpers
// ---------------------------------------------------------------------------
static void gemm(hipStream_t st, const float* A, const float* B, const float* bias,
                 const float* C, float* Dst, int M, int N, int K, int lda, int ldb,
                 int ldc, int ldd, int tA, int tB, float alpha, float beta,
                 int nbat = 1, int Hb = 1, long long sAb = 0, long long sAh = 0,
                 long long sBb = 0, long long sBh = 0, long long sCb = 0,
                 long long sCh = 0, long long sDb = 0, long long sDh = 0) {
  GemmP p;
  p.A = A; p.B = B; p.bias = bias; p.C = C; p.Dst = Dst;
  p.sAb = sAb; p.sAh = sAh; p.sBb = sBb; p.sBh = sBh;
  p.sCb = sCb; p.sCh = sCh; p.sDb = sDb; p.sDh = sDh;
  p.M = M; p.N = N; p.K = K;
  p.lda = lda; p.ldb = ldb; p.ldc = ldc; p.ldd = ldd;
  p.transA = tA; p.transB = tB; p.Hb = Hb;
  p.alpha = alpha; p.beta = beta;
  dim3 grid(N / 64, M / 64, nbat);
  k_gemm<<<grid, dim3(128), 0, st>>>(p);
}

static inline int ew_blocks(int n) {
  int b = (n + 255) / 256;
  return b > 4096 ? 4096 : b;
}

extern "C" void kernel_launch(void* const* d_in, const int* in_sizes, int n_in,
                              void* d_out, int out_size, void* d_ws, size_t ws_size,
                              hipStream_t stream) {
  const int NB = 4, D = 512, DFF = 2048, V = 32000, H = 8, Bb = 2, S = 512, T = 4;
  const int M = Bb * S;      // 1024 rows for all [B,S,*] mats
  const int Dh = D / H;      // 64
  const float LR = 0.05f;
  const float ISQ = 0.125f;  // 1/sqrt(dh)

  const int*   input_ids  = (const int*)d_in[0];
  const int*   target_ids = (const int*)d_in[1];
  const float* word_emb   = (const float*)d_in[2];
  const float* pos_emb    = (const float*)d_in[3];
  const float* Wq = (const float*)d_in[4];
  const float* Wk = (const float*)d_in[5];
  const float* Wv = (const float*)d_in[6];
  const float* bq = (const float*)d_in[7];
  const float* bk = (const float*)d_in[8];
  const float* bv = (const float*)d_in[9];
  const float* Wo = (const float*)d_in[10];
  const float* bo = (const float*)d_in[11];
  const float* W1 = (const float*)d_in[12];
  const float* b1 = (const float*)d_in[13];
  const float* W2 = (const float*)d_in[14];
  const float* b2 = (const float*)d_in[15];
  const float* Wout = (const float*)d_in[16];
  const float* bout = (const float*)d_in[17];
  const float* xq_in  = (const float*)d_in[18];
  const float* xao_in = (const float*)d_in[19];
  const float* xf1_in = (const float*)d_in[20];
  const float* xf2_in = (const float*)d_in[21];
  const float* xo_in  = (const float*)d_in[22];
  float* out = (float*)d_out;

  // ---- workspace carve (floats) ----
  const size_t SZ_XQ  = (size_t)NB * M * D;    // 2,097,152
  const size_t SZ_XF2 = (size_t)NB * M * DFF;  // 8,388,608
  const size_t SZ_XO  = (size_t)M * D;         //   524,288
  const size_t SZ_BSD = (size_t)M * D;
  const size_t SZ_P   = (size_t)Bb * H * S * S;
  const size_t SZ_U   = (size_t)M * DFF;

  float* w = (float*)d_ws;
  size_t off = 0;
  float* s_xq[2];  s_xq[0] = w + off; off += SZ_XQ;  s_xq[1] = w + off; off += SZ_XQ;
  float* s_xao[2]; s_xao[0] = w + off; off += SZ_XQ; s_xao[1] = w + off; off += SZ_XQ;
  float* s_xf1[2]; s_xf1[0] = w + off; off += SZ_XQ; s_xf1[1] = w + off; off += SZ_XQ;
  float* s_xf2[2]; s_xf2[0] = w + off; off += SZ_XF2; s_xf2[1] = w + off; off += SZ_XF2;
  float* s_xo[2];  s_xo[0] = w + off; off += SZ_XO;  s_xo[1] = w + off; off += SZ_XO;
  float* Qb    = w + off; off += SZ_BSD;
  float* Kb    = w + off; off += SZ_BSD;
  float* Vb    = w + off; off += SZ_BSD;
  float* Ea    = w + off; off += SZ_BSD;   // eps_a (= dO of attention VJP)
  float* dQ    = w + off; off += SZ_BSD;
  float* dK    = w + off; off += SZ_BSD;
  float* dV    = w + off; off += SZ_BSD;
  float* predA = w + off; off += SZ_BSD;
  float* predL = w + off; off += SZ_BSD;
  float* epsL  = w + off; off += SZ_BSD;
  float* Pb    = w + off; off += SZ_P;     // attention probs
  float* dPb   = w + off; off += SZ_P;     // dP, then G (in place)
  float* u1    = w + off; off += SZ_U;     // fc1 pre-activation
  float* g1    = w + off; off += SZ_U;     // eps1 * gelu'(u1)
  float* ws_en = w + off; off += 16;       // per-step energies

  // batch strides for per-head views of [B,S,D] and of [B,H,S,S]
  const long long sBD_b = (long long)S * D, sBD_h = Dh;
  const long long sP_b  = (long long)H * S * S, sP_h = (long long)S * S;
  const int nbat = Bb * H;

  k_zero<<<1, 16, 0, stream>>>(ws_en, 16);

  const float* c_xq  = xq_in;
  const float* c_xao = xao_in;
  const float* c_xf1 = xf1_in;
  const float* c_xf2 = xf2_in;
  const float* c_xo  = xo_in;

  for (int t = 0; t < T; ++t) {
    float* en = ws_en + t;
    float* n_xq  = s_xq[t & 1];
    float* n_xao = s_xao[t & 1];
    float* n_xf1 = s_xf1[t & 1];
    float* n_xf2 = s_xf2[t & 1];
    float* n_xo  = s_xo[t & 1];

    // ---- output PC layer: pred_out -> eps_out (in d_out) -> xo update ----
    gemm(stream, c_xo, Wout, bout, nullptr, out, M, V, D, D, D, 0, V, 0, 1, 1.f, 0.f);
    k_eps_out<<<dim3(V / 256, M), 256, 0, stream>>>(out, target_ids, en, V);
    gemm(stream, out, Wout, nullptr, c_xo, n_xo, M, D, V, V, D, D, D, 0, 0, LR, 1.f);

    // ---- embedding PC layer energy ----
    k_eps_embed<<<ew_blocks(M * D), 256, 0, stream>>>(c_xq, input_ids, word_emb,
                                                      pos_emb, en, Bb, S, D);

    // ---- per-block PC updates ----
    for (int nb = 0; nb < NB; ++nb) {
      const float* xqb  = c_xq  + (size_t)nb * M * D;
      const float* xaob = c_xao + (size_t)nb * M * D;
      const float* xf1b = c_xf1 + (size_t)nb * M * D;
      const float* xf2b = c_xf2 + (size_t)nb * M * DFF;
      const float* tgt2 = (nb < NB - 1) ? (c_xq + (size_t)(nb + 1) * M * D) : c_xo;
      float* nxqb  = n_xq  + (size_t)nb * M * D;
      float* nxaob = n_xao + (size_t)nb * M * D;
      float* nxf1b = n_xf1 + (size_t)nb * M * D;
      float* nxf2b = n_xf2 + (size_t)nb * M * DFF;
      const float* Wqb = Wq + (size_t)nb * D * D;
      const float* Wkb = Wk + (size_t)nb * D * D;
      const float* Wvb = Wv + (size_t)nb * D * D;
      const float* Wob = Wo + (size_t)nb * D * D;
      const float* W1b = W1 + (size_t)nb * DFF * D;
      const float* W2b = W2 + (size_t)nb * D * DFF;

      // QKV projections
      gemm(stream, xqb, Wqb, bq + nb * D, nullptr, Qb, M, D, D, D, D, 0, D, 0, 1, 1.f, 0.f);
      gemm(stream, xqb, Wkb, bk + nb * D, nullptr, Kb, M, D, D, D, D, 0, D, 0, 1, 1.f, 0.f);
      gemm(stream, xqb, Wvb, bv + nb * D, nullptr, Vb, M, D, D, D, D, 0, D, 0, 1, 1.f, 0.f);
      // scores = Q K^T / sqrt(dh)  (batched over B*H)
      gemm(stream, Qb, Kb, nullptr, nullptr, Pb, S, S, Dh, D, D, 0, S, 0, 1, ISQ, 0.f,
           nbat, H, sBD_b, sBD_h, sBD_b, sBD_h, 0, 0, sP_b, sP_h);
      k_softmax_causal<<<Bb * H * S, 256, 0, stream>>>(Pb, S);
      // pred_a = P V
      gemm(stream, Pb, Vb, nullptr, nullptr, predA, S, Dh, S, S, D, 0, D, 0, 0, 1.f, 0.f,
           nbat, H, sP_b, sP_h, sBD_b, sBD_h, 0, 0, sBD_b, sBD_h);
      // eps_a
      k_diff_energy<<<ew_blocks(M * D), 256, 0, stream>>>(xaob, predA, Ea, en, M * D);
      // attention VJP
      gemm(stream, Ea, Vb, nullptr, nullptr, dPb, S, S, Dh, D, D, 0, S, 0, 1, 1.f, 0.f,
           nbat, H, sBD_b, sBD_h, sBD_b, sBD_h, 0, 0, sP_b, sP_h);           // dP = dO V^T
      gemm(stream, Pb, Ea, nullptr, nullptr, dV, S, Dh, S, S, D, 0, D, 1, 0, 1.f, 0.f,
           nbat, H, sP_b, sP_h, sBD_b, sBD_h, 0, 0, sBD_b, sBD_h);           // dV = P^T dO
      k_softmax_bwd<<<Bb * H * S, 256, 0, stream>>>(dPb, Pb, ISQ, S);        // G (in dPb)
      gemm(stream, dPb, Kb, nullptr, nullptr, dQ, S, Dh, S, S, D, 0, D, 0, 0, 1.f, 0.f,
           nbat, H, sP_b, sP_h, sBD_b, sBD_h, 0, 0, sBD_b, sBD_h);           // dQ = G K
      gemm(stream, dPb, Qb, nullptr, nullptr, dK, S, Dh, S, S, D, 0, D, 1, 0, 1.f, 0.f,
           nbat, H, sP_b, sP_h, sBD_b, sBD_h, 0, 0, sBD_b, sBD_h);           // dK = G^T Q
      // xq_n = xq + LR*(dQ Wq + dK Wk + dV Wv)
      gemm(stream, dQ, Wqb, nullptr, xqb, nxqb, M, D, D, D, D, D, D, 0, 0, LR, 1.f);
      gemm(stream, dK, Wkb, nullptr, nxqb, nxqb, M, D, D, D, D, D, D, 0, 0, LR, 1.f);
      gemm(stream, dV, Wvb, nullptr, nxqb, nxqb, M, D, D, D, D, D, D, 0, 0, LR, 1.f);

      // pc_output (linear)
      gemm(stream, xaob, Wob, bo + nb * D, nullptr, predL, M, D, D, D, D, 0, D, 0, 1, 1.f, 0.f);
      k_diff_energy<<<ew_blocks(M * D), 256, 0, stream>>>(xf1b, predL, epsL, en, M * D);
      gemm(stream, epsL, Wob, nullptr, xaob, nxaob, M, D, D, D, D, D, D, 0, 0, LR, 1.f);

      // pc_layer1 (fc1 + exact GELU)
      gemm(stream, xf1b, W1b, b1 + nb * DFF, nullptr, u1, M, DFF, D, D, D, 0, DFF, 0, 1, 1.f, 0.f);
      k_gelu_bwd<<<ew_blocks(M * DFF), 256, 0, stream>>>(u1, xf2b, g1, en, M * DFF);
      gemm(stream, g1, W1b, nullptr, xf1b, nxf1b, M, D, DFF, DFF, D, D, D, 0, 0, LR, 1.f);

      // pc_layer2 (linear), target = next block's x_qkv (or x_out)
      gemm(stream, xf2b, W2b, b2 + nb * D, nullptr, predL, M, D, DFF, DFF, DFF, 0, D, 0, 1, 1.f, 0.f);
      k_diff_energy<<<ew_blocks(M * D), 256, 0, stream>>>(tgt2, predL, epsL, en, M * D);
      gemm(stream, epsL, W2b, nullptr, xf2b, nxf2b, M, DFF, D, D, DFF, DFF, DFF, 0, 0, LR, 1.f);
    }

    c_xq = n_xq; c_xao = n_xao; c_xf1 = n_xf1; c_xf2 = n_xf2; c_xo = n_xo;
  }

  // ---- final logits + last-step energy ----
  gemm(stream, c_xo, Wout, bout, nullptr, out, M, V, D, D, D, 0, V, 0, 1, 1.f, 0.f);
  k_copy1<<<1, 32, 0, stream>>>(out + (size_t)M * V, ws_en + (T - 1));
}